// PlainHead_44839458570506
// MI455X (gfx1250) — compile-verified
//
#include <hip/hip_runtime.h>
#include <hip/hip_bf16.h>

#define B_DIM 32
#define C_DIM 128
#define N_DIM 65536
#define K_TOP 6553   // max(int(65536*0.1), 1)

typedef __attribute__((ext_vector_type(2))) float v2f;
typedef __attribute__((ext_vector_type(8))) float v8f;

// -----------------------------------------------------------------------------
// Kernel 1: scores[b][n] = sum_c x[b][c][n] * W[c] + bias
//
// One wave computes 32 consecutive n values with TWO interleaved accumulators:
// WMMA tile A owns memory columns n0+2j, tile B owns n0+2j+1 (j = 0..15).
// Per C-quad step each lane does two global_load_b64 (one float2 per x row,
// feeding both tiles), then two chained V_WMMA_F32_16X16X4_F32. A-matrix rows
// are the replicated weight quad, so every D row equals the score row; lane j
// (j<16) reads D[0][j] from acc VGPR0 and stores a contiguous float2.
// K-index lane mapping per ISA 7.12.2: lanes 0-15 carry K=0/1 (VGPR0/1),
// lanes 16-31 carry K=2/3 — applied identically to A and B operands.
// Each x element is read exactly once in full-128B half-wave requests:
// HBM-bound, ~1.07 GB @ 23.3 TB/s ≈ 46 us floor.
// -----------------------------------------------------------------------------
__global__ __launch_bounds__(128)
void score_wmma_kernel(const float* __restrict__ x, const float* __restrict__ W,
                       const float* __restrict__ bias, float* __restrict__ scores) {
    const int lane  = threadIdx.x & 31;
    const int gwave = blockIdx.x * 4 + (threadIdx.x >> 5);
    const int b     = gwave >> 11;                 // 2048 tiles of 32 n per batch
    const int n0    = (gwave & 2047) * 32;
    const int col   = lane & 15;
    const int khalf = (lane >> 4) * 2;             // 0 for lanes 0-15, 2 for 16-31

    // per-lane base: batch b, column n0 + 2*col (float2 covers both tiles)
    const size_t xbase = (size_t)b * C_DIM * N_DIM + (size_t)n0 + 2 * col;

    v8f accA = {};
    v8f accB = {};
    for (int c0 = 0; c0 < C_DIM; c0 += 4) {
        v2f a;
        a.x = W[c0 + khalf];
        a.y = W[c0 + khalf + 1];
        // row c0+khalf and row c0+khalf+1, two adjacent n each (b64 loads)
        v2f xr0 = *(const v2f*)(x + xbase + (size_t)(c0 + khalf)     * N_DIM);
        v2f xr1 = *(const v2f*)(x + xbase + (size_t)(c0 + khalf + 1) * N_DIM);
        v2f bm0; bm0.x = xr0.x; bm0.y = xr1.x;     // columns n0 + 2j
        v2f bm1; bm1.x = xr0.y; bm1.y = xr1.y;     // columns n0 + 2j + 1
        // 8 args: (neg_a, A, neg_b, B, c_mod, C, reuse_a, reuse_b)
        accA = __builtin_amdgcn_wmma_f32_16x16x4_f32(false, a, false, bm0,
                                                     (short)0, accA, false, false);
        accB = __builtin_amdgcn_wmma_f32_16x16x4_f32(false, a, false, bm1,
                                                     (short)0, accB, false, false);
    }
    if (lane < 16) {
        v2f r;
        r.x = accA[0] + bias[0];
        r.y = accB[0] + bias[0];
        *(v2f*)(scores + (size_t)b * N_DIM + n0 + 2 * col) = r;
    }
}

// -----------------------------------------------------------------------------
// Kernel 2: exact top-k(|scores|) mean per batch. One 1024-thread block per
// batch; scores (256 KB/batch, 8 MB total) are L2-resident after kernel 1
// (global L2 = 192 MB). |score| >= 0 so its IEEE bit pattern is monotonic ->
// 3-level radix select: bits [30:20] (2048 bins) -> bits [19:9] (2048 bins)
// -> bits [8:0] (512 bins) gives the exact 32-bit k-th value T and the number
// of ties to include. Final pass: sum of values > T, plus ties*T, over k.
// -----------------------------------------------------------------------------
__global__ __launch_bounds__(1024)
void topk_mean_kernel(const float* __restrict__ scores, float* __restrict__ out) {
    __shared__ unsigned hist[2048];
    __shared__ float    red[1024];
    __shared__ unsigned s_t, s_r;

    const int    tid  = threadIdx.x;
    const int    b    = blockIdx.x;
    const size_t base = (size_t)b * N_DIM;

    // ---- level 1: bits [30:20] ----
    for (int i = tid; i < 2048; i += 1024) hist[i] = 0;
    __syncthreads();
    for (int i = tid; i < N_DIM; i += 1024) {
        unsigned u = __float_as_uint(fabsf(scores[base + i]));
        atomicAdd(&hist[u >> 20], 1u);
    }
    __syncthreads();
    if (tid == 0) {
        unsigned cum = 0; int t = 2047;
        for (; t > 0; --t) { if (cum + hist[t] >= (unsigned)K_TOP) break; cum += hist[t]; }
        s_t = (unsigned)t; s_r = (unsigned)K_TOP - cum;
    }
    __syncthreads();
    const unsigned t1 = s_t, r1 = s_r;

    // ---- level 2: bits [19:9] within bin t1 ----
    for (int i = tid; i < 2048; i += 1024) hist[i] = 0;
    __syncthreads();
    for (int i = tid; i < N_DIM; i += 1024) {
        unsigned u = __float_as_uint(fabsf(scores[base + i]));
        if ((u >> 20) == t1) atomicAdd(&hist[(u >> 9) & 0x7FF], 1u);
    }
    __syncthreads();
    if (tid == 0) {
        unsigned cum = 0; int t = 2047;
        for (; t > 0; --t) { if (cum + hist[t] >= r1) break; cum += hist[t]; }
        s_t = (unsigned)t; s_r = r1 - cum;
    }
    __syncthreads();
    const unsigned t2 = s_t, r2 = s_r;
    const unsigned pref = (t1 << 11) | t2;          // value of (u >> 9)

    // ---- level 3: bits [8:0] within sub-bin ----
    for (int i = tid; i < 512; i += 1024) hist[i] = 0;
    __syncthreads();
    for (int i = tid; i < N_DIM; i += 1024) {
        unsigned u = __float_as_uint(fabsf(scores[base + i]));
        if ((u >> 9) == pref) atomicAdd(&hist[u & 0x1FF], 1u);
    }
    __syncthreads();
    if (tid == 0) {
        unsigned cum = 0; int t = 511;
        for (; t > 0; --t) { if (cum + hist[t] >= r2) break; cum += hist[t]; }
        s_t = (unsigned)t; s_r = r2 - cum;          // ties at threshold to include
    }
    __syncthreads();
    const unsigned uT    = (pref << 9) | s_t;       // exact k-th |score| bit pattern
    const unsigned nties = s_r;
    const float    T     = __uint_as_float(uT);

    // ---- final pass: sum of strictly-greater values ----
    float psum = 0.f;
    for (int i = tid; i < N_DIM; i += 1024) {
        unsigned u = __float_as_uint(fabsf(scores[base + i]));
        if (u > uT) psum += __uint_as_float(u);
    }
    red[tid] = psum;
    __syncthreads();
    for (int s = 512; s > 0; s >>= 1) {
        if (tid < s) red[tid] += red[tid + s];
        __syncthreads();
    }
    if (tid == 0) out[b] = (red[0] + (float)nties * T) / (float)K_TOP;
}

extern "C" void kernel_launch(void* const* d_in, const int* in_sizes, int n_in,
                              void* d_out, int out_size, void* d_ws, size_t ws_size,
                              hipStream_t stream) {
    const float* x    = (const float*)d_in[0];   // [B, C, N] fp32
    const float* W    = (const float*)d_in[1];   // [C] fp32
    const float* bias = (const float*)d_in[2];   // [1] fp32
    float*       out  = (float*)d_out;           // [B] fp32
    float*   scores   = (float*)d_ws;            // B*N floats = 8 MB scratch

    const int nblocks = (B_DIM * (N_DIM / 32)) / 4;   // 4 waves per 128-thr block
    hipLaunchKernelGGL(score_wmma_kernel, dim3(nblocks), dim3(128), 0, stream,
                       x, W, bias, scores);
    hipLaunchKernelGGL(topk_mean_kernel, dim3(B_DIM), dim3(1024), 0, stream,
                       scores, out);
}